// S4Layer_43198781063516
// MI455X (gfx1250) — compile-verified
//
#include <hip/hip_runtime.h>
#include <hip/hip_bf16.h>

// Problem constants (match reference)
#define BATCH 8
#define SEQ   4096
#define HD    512      // H
#define NS    64       // N states
#define CHUNK 64       // chunk length
#define NCH   64       // SEQ / CHUNK
#define KDIM  1024     // 2H

typedef __attribute__((ext_vector_type(2))) float v2f;
typedef __attribute__((ext_vector_type(8))) float v8f;

// ---------------------------------------------------------------------------
// Kernel 1: discretization parameters for both directions
//   Af/Ar  = A_bar, Bbf/Bbr = B_bar, Adf/Adr = A_bar^CHUNK
// ---------------------------------------------------------------------------
__global__ __launch_bounds__(256) void k_precompute(
    const float* __restrict__ A_log, const float* __restrict__ Bv,
    const float* __restrict__ A_log_r, const float* __restrict__ B_r,
    const float* __restrict__ log_dt,
    float* __restrict__ Af, float* __restrict__ Bbf, float* __restrict__ Adf,
    float* __restrict__ Ar, float* __restrict__ Bbr, float* __restrict__ Adr) {
  int idx = blockIdx.x * blockDim.x + threadIdx.x;   // < HD*NS
  int h = idx >> 6;
  float dt = __expf(log_dt[h]);
  {
    float eA = __expf(A_log[idx]);
    float ab = __expf(-dt * eA);
    Af[idx]  = ab;
    Bbf[idx] = (ab - 1.0f) / (-eA) * Bv[idx];
    float la = __logf(fmaxf(ab, 1e-30f));
    Adf[idx] = __expf((float)CHUNK * la);
  }
  {
    float eA = __expf(A_log_r[idx]);
    float ab = __expf(-dt * eA);
    Ar[idx]  = ab;
    Bbr[idx] = (ab - 1.0f) / (-eA) * B_r[idx];
    float la = __logf(fmaxf(ab, 1e-30f));
    Adr[idx] = __expf((float)CHUNK * la);
  }
}

// ---------------------------------------------------------------------------
// Kernel 2: build per-(h,dir) semiseparable block matrices (64x64 each):
//   M[j,k] = K[j-k] for k<=j (K[d] = sum_n C_n*Bb_n*A_n^d), else 0
//   P[n,j] = A_n^(63-j) * Bb_n          (chunk-local end-state operator)
//   Q[j,n] = C_n * A_n^(j+1)            (carry-in correction operator)
// grid (HD, 2), block 64
// ---------------------------------------------------------------------------
__global__ __launch_bounds__(64) void k_build(
    const float* __restrict__ Af, const float* __restrict__ Bbf,
    const float* __restrict__ Ar, const float* __restrict__ Bbr,
    const float* __restrict__ Cf, const float* __restrict__ Cr,
    float* __restrict__ Mg, float* __restrict__ Pg, float* __restrict__ Qg) {
  const int h = blockIdx.x, dir = blockIdx.y;
  const int t = threadIdx.x;                    // 0..63
  __shared__ float Apow[NS * (NS + 1)];         // Apow[n*65+d] = A_n^d
  __shared__ float A_s[NS], Bb_s[NS], C_s[NS], K_s[NS];

  const float* Ap  = dir ? Ar  : Af;
  const float* Bbp = dir ? Bbr : Bbf;
  const float* Cp  = dir ? Cr  : Cf;
  float a  = Ap[h * NS + t];
  float bb = Bbp[h * NS + t];
  float cc = Cp[h * NS + t];
  A_s[t] = a; Bb_s[t] = bb; C_s[t] = cc;
  float p = 1.0f;
  for (int d = 0; d < NS; ++d) { Apow[t * 65 + d] = p; p *= a; }
  __syncthreads();

  // K[t] = sum_n C_n * Bb_n * A_n^t
  float kd = 0.0f;
  for (int n = 0; n < NS; ++n) kd = fmaf(C_s[n] * Bb_s[n], Apow[n * 65 + t], kd);
  K_s[t] = kd;
  __syncthreads();

  const size_t off = ((size_t)h * 2 + dir) * (NS * NS);
  float* Mrow = Mg + off + t * NS;              // row j = t
  float* Prow = Pg + off + t * NS;              // row n = t
  float* Qrow = Qg + off + t * NS;              // row j = t
  for (int k = 0; k < NS; ++k) Mrow[k] = (k <= t) ? K_s[t - k] : 0.0f;
  for (int j = 0; j < NS; ++j) Prow[j] = Apow[t * 65 + (63 - j)] * bb;
  for (int n = 0; n < NS; ++n) Qrow[n] = C_s[n] * Apow[n * 65 + t] * A_s[n];
}

// ---------------------------------------------------------------------------
// Kernel 3: u [B,L,H] -> uT [B,H,L]  (coalesced both sides via LDS tile)
// ---------------------------------------------------------------------------
__global__ __launch_bounds__(256) void k_transpose_u(
    const float* __restrict__ u, float* __restrict__ uT) {
  __shared__ float tile[32][33];
  const int b  = blockIdx.z;
  const int l0 = blockIdx.x * 32;
  const int h0 = blockIdx.y * 32;
  const int tx = threadIdx.x, ty = threadIdx.y;      // (32, 8)
  const float* ub = u + (size_t)b * SEQ * HD;
  float* uTb = uT + (size_t)b * HD * SEQ;
#pragma unroll
  for (int r = 0; r < 32; r += 8)
    tile[r + ty][tx] = ub[(size_t)(l0 + r + ty) * HD + (h0 + tx)];
  __syncthreads();
#pragma unroll
  for (int r = 0; r < 32; r += 8)
    uTb[(size_t)(h0 + r + ty) * SEQ + (l0 + tx)] = tile[tx][r + ty];
}

// ---------------------------------------------------------------------------
// Kernel 4: WMMA chunked scan. One block per (h, dir, b); 256 threads = 8 waves.
//   Phase A: locals(64n x 64c) = P @ U          (WMMA, U from LDS)
//   Phase B: carry scan over chunks per state   (64 lanes, 64 steps)
//   Phase C: Y = M @ U + Q @ S_in (+ D*u)       (WMMA), scatter into ycat
//   U[j,c] = u_s[c*64+j]  (flipped sequence for dir=1)
// ---------------------------------------------------------------------------
__global__ __launch_bounds__(256) void k_chunk(
    const float* __restrict__ uT,
    const float* __restrict__ Mg, const float* __restrict__ Pg, const float* __restrict__ Qg,
    const float* __restrict__ Adf, const float* __restrict__ Adr,
    const float* __restrict__ D, float* __restrict__ ycat) {
  const int h = blockIdx.x, dir = blockIdx.y, b = blockIdx.z;
  const int tid  = threadIdx.x;
  const int lane = tid & 31;
  const int w    = tid >> 5;                    // wave 0..7

  __shared__ float u_s[SEQ];                    // 16 KB (possibly flipped)
  __shared__ float loc_s[NS * 65];              // locals[n*65+c]   (b32 access)
  __shared__ float sin_s[NCH * 66];             // carries[c*66+n]  (b64-aligned reads)

  const float* up = uT + ((size_t)b * HD + h) * SEQ;
  if (dir == 0) {
    for (int i = tid; i < SEQ; i += 256) u_s[i] = up[i];
  } else {
    for (int i = tid; i < SEQ; i += 256) u_s[i] = up[SEQ - 1 - i];
  }
  __syncthreads();

  const size_t off = ((size_t)h * 2 + dir) * (NS * NS);
  const float* Mm = Mg + off;
  const float* Pm = Pg + off;
  const float* Qm = Qg + off;
  const int r16   = lane & 15;
  const int kb    = (lane >> 4) * 2;
  const int col   = lane & 15;
  const int rhalf = (lane >> 4) * 8;

  // ---- Phase A: locals = P @ U ----
  for (int ti = w; ti < 16; ti += 8) {
    const int m0  = (ti >> 2) * 16;             // state rows n
    const int n0c = (ti & 3) * 16;              // chunk cols c
    v8f acc = {};
    const float* ap = Pm + (m0 + r16) * NS + kb;
    const int bbase = (n0c + r16) * CHUNK + kb;
#pragma unroll 4
    for (int k = 0; k < NS; k += 4) {
      v2f a = *(const v2f*)(ap + k);
      v2f bf = *(const v2f*)(&u_s[bbase + k]);
      acc = __builtin_amdgcn_wmma_f32_16x16x4_f32(false, a, false, bf, (short)0, acc, false, false);
    }
#pragma unroll
    for (int r = 0; r < 8; ++r)
      loc_s[(m0 + r + rhalf) * 65 + (n0c + col)] = acc[r];
  }
  __syncthreads();

  // ---- Phase B: inter-chunk exclusive carry scan (lane t == state n) ----
  if (tid < NS) {
    const float* Adp = dir ? Adr : Adf;
    float ad = Adp[h * NS + tid];
    float carry = 0.0f;
    for (int c = 0; c < NCH; ++c) {
      float tmp = loc_s[tid * 65 + c];
      sin_s[c * 66 + tid] = carry;              // state entering chunk c
      carry = fmaf(ad, carry, tmp);             // S_c = A^64 * S_{c-1} + local_c
    }
  }
  __syncthreads();

  // ---- Phase C: Y = M @ U + Q @ S_in, add D*u, scatter to ycat ----
  const float Dh = D[h];
  for (int ti = w; ti < 16; ti += 8) {
    const int m0  = (ti >> 2) * 16;             // time rows j
    const int n0c = (ti & 3) * 16;              // chunk cols c
    v8f acc = {};
    const float* am = Mm + (m0 + r16) * NS + kb;
    const int bbase = (n0c + r16) * CHUNK + kb;
#pragma unroll 4
    for (int k = 0; k < NS; k += 4) {
      v2f a = *(const v2f*)(am + k);
      v2f bf = *(const v2f*)(&u_s[bbase + k]);
      acc = __builtin_amdgcn_wmma_f32_16x16x4_f32(false, a, false, bf, (short)0, acc, false, false);
    }
    const float* aq = Qm + (m0 + r16) * NS + kb;
    const int sbase = (n0c + r16) * 66 + kb;
#pragma unroll 4
    for (int k = 0; k < NS; k += 4) {
      v2f a = *(const v2f*)(aq + k);
      v2f bf = *(const v2f*)(&sin_s[sbase + k]);
      acc = __builtin_amdgcn_wmma_f32_16x16x4_f32(false, a, false, bf, (short)0, acc, false, false);
    }
#pragma unroll
    for (int r = 0; r < 8; ++r) {
      int j = m0 + r + rhalf;
      int c = n0c + col;
      int li = c * CHUNK + j;
      float val = acc[r] + Dh * u_s[li];
      int lg = dir ? (SEQ - 1 - li) : li;
      ycat[((size_t)b * SEQ + lg) * KDIM + dir * HD + h] = val;   // lands in L2
    }
  }
}

// ---------------------------------------------------------------------------
// Kernel 5: out[m, h] = sum_k ycat[m, k] * W_bi[h, k] + b_bi[h]
//   M = 32768, N = 512, K = 1024. fp32 WMMA, 32x32 tile per wave.
// ---------------------------------------------------------------------------
__global__ __launch_bounds__(256) void k_gemm_wmma(
    const float* __restrict__ ycat, const float* __restrict__ W,
    const float* __restrict__ bias, float* __restrict__ out) {
  const int lane = threadIdx.x & 31;
  const int wid  = blockIdx.x * (blockDim.x >> 5) + (threadIdx.x >> 5);
  const int tileM = wid >> 4;
  const int tileN = wid & 15;
  const int m0 = tileM * 32;
  const int n0 = tileN * 32;
  const int r16 = lane & 15;
  const int kb  = (lane >> 4) * 2;

  const float* a0p = ycat + (size_t)(m0 + r16) * KDIM + kb;
  const float* a1p = a0p + (size_t)16 * KDIM;
  const float* b0p = W + (size_t)(n0 + r16) * KDIM + kb;
  const float* b1p = b0p + (size_t)16 * KDIM;

  v8f c00 = {}; v8f c01 = {}; v8f c10 = {}; v8f c11 = {};
  for (int k = 0; k < KDIM; k += 4) {
    __builtin_prefetch(a0p + k + 256, 0, 1);
    v2f a0 = *(const v2f*)(a0p + k);
    v2f a1 = *(const v2f*)(a1p + k);
    v2f b0 = *(const v2f*)(b0p + k);
    v2f b1 = *(const v2f*)(b1p + k);
    c00 = __builtin_amdgcn_wmma_f32_16x16x4_f32(false, a0, false, b0, (short)0, c00, false, false);
    c01 = __builtin_amdgcn_wmma_f32_16x16x4_f32(false, a0, false, b1, (short)0, c01, false, false);
    c10 = __builtin_amdgcn_wmma_f32_16x16x4_f32(false, a1, false, b0, (short)0, c10, false, false);
    c11 = __builtin_amdgcn_wmma_f32_16x16x4_f32(false, a1, false, b1, (short)0, c11, false, false);
  }

  const int col   = lane & 15;
  const int rhalf = (lane >> 4) * 8;
  const float bias0 = bias[n0 + col];
  const float bias1 = bias[n0 + 16 + col];
#pragma unroll
  for (int r = 0; r < 8; ++r) {
    int row = r + rhalf;
    out[(size_t)(m0 + row) * HD + (n0 + col)]           = c00[r] + bias0;
    out[(size_t)(m0 + row) * HD + (n0 + 16 + col)]      = c01[r] + bias1;
    out[(size_t)(m0 + 16 + row) * HD + (n0 + col)]      = c10[r] + bias0;
    out[(size_t)(m0 + 16 + row) * HD + (n0 + 16 + col)] = c11[r] + bias1;
  }
}

// ---------------------------------------------------------------------------
extern "C" void kernel_launch(void* const* d_in, const int* in_sizes, int n_in,
                              void* d_out, int out_size, void* d_ws, size_t ws_size,
                              hipStream_t stream) {
  (void)in_sizes; (void)n_in; (void)out_size; (void)ws_size;
  const float* u       = (const float*)d_in[0];
  const float* A_log   = (const float*)d_in[1];
  const float* Bv      = (const float*)d_in[2];
  const float* C       = (const float*)d_in[3];
  const float* D       = (const float*)d_in[4];
  const float* log_dt  = (const float*)d_in[5];
  const float* A_log_r = (const float*)d_in[6];
  const float* B_r     = (const float*)d_in[7];
  const float* C_r     = (const float*)d_in[8];
  const float* W_bi    = (const float*)d_in[9];
  const float* b_bi    = (const float*)d_in[10];
  float* out = (float*)d_out;

  float* ws   = (float*)d_ws;
  float* uT   = ws;                                        // B*H*L
  float* ycat = uT + (size_t)BATCH * HD * SEQ;             // B*L*2H
  float* Af   = ycat + (size_t)BATCH * SEQ * KDIM;
  float* Bbf  = Af  + HD * NS;
  float* Adf  = Bbf + HD * NS;
  float* Ar   = Adf + HD * NS;
  float* Bbr  = Ar  + HD * NS;
  float* Adr  = Bbr + HD * NS;
  float* Mg   = Adr + HD * NS;                             // H*2*64*64 each
  float* Pg   = Mg  + (size_t)HD * 2 * NS * NS;
  float* Qg   = Pg  + (size_t)HD * 2 * NS * NS;

  // 1) discretization parameters
  k_precompute<<<(HD * NS) / 256, 256, 0, stream>>>(
      A_log, Bv, A_log_r, B_r, log_dt, Af, Bbf, Adf, Ar, Bbr, Adr);

  // 2) semiseparable block operators M, P, Q per (h, dir)
  k_build<<<dim3(HD, 2), 64, 0, stream>>>(Af, Bbf, Ar, Bbr, C, C_r, Mg, Pg, Qg);

  // 3) transpose u -> uT [B,H,L]
  k_transpose_u<<<dim3(SEQ / 32, HD / 32, BATCH), dim3(32, 8), 0, stream>>>(u, uT);

  // 4) WMMA chunked scan -> ycat [B,L,2H]
  k_chunk<<<dim3(HD, 2, BATCH), 256, 0, stream>>>(
      uT, Mg, Pg, Qg, Adf, Adr, D, ycat);

  // 5) fp32 WMMA output projection -> out [B*L, H]
  k_gemm_wmma<<<2048, 256, 0, stream>>>(ycat, W_bi, b_bi, out);
}